// GlobalPointer_84482006713191
// MI455X (gfx1250) — compile-verified
//
#include <hip/hip_runtime.h>
#include <hip/hip_bf16.h>
#include <cstddef>

// ---------------------------------------------------------------------------
// GlobalPointer NER head for MI455X (gfx1250), fp32 WMMA path.
// Problem sizes (fixed by reference): B=16, S=512, H=768, NER=9, d=64.
//
// Roofline: output logits = 151 MB fp32 -> ~6.5 us at 23.3 TB/s; total FLOPs
// ~20 GFLOP fp32, trivial for the WMMA units. So: fp32 WMMA (16x16x4) keeps
// reference numerics, and the optimization budget goes to write coalescing
// (LDS-restaged float4 stores) and load overlap (global_prefetch).
// ---------------------------------------------------------------------------

#define B_SZ    16
#define S_LEN   512
#define H_DIM   768
#define NER     9
#define D_INNER 64
#define N_OUT   (NER * 2 * D_INNER)   // 1152
#define NEG_BIG 1000000000000.0f

typedef __attribute__((ext_vector_type(2))) float v2f;
typedef __attribute__((ext_vector_type(8))) float v8f;

// LDS row stride (floats): 68 = 64 + 4 pad. Keeps float4 accesses 16B-aligned
// (68*4 = 272 bytes) and spreads the 16-row column reads across banks.
#define LDS_STRIDE 68

// fp32 WMMA 16x16x4, ISA layout:
//   A (16x4, MxK): lane L holds row m = L%16; VGPR0/1 = k = koff, koff+1
//                  where koff = (L<16) ? 0 : 2.
//   B (4x16, KxN): symmetric when B is stored as [n][k] row-major.
//   C/D (16x16):   VGPR r, lanes 0-15 -> (m=r,   n=L);
//                            lanes16-31 -> (m=r+8, n=L-16).

// ---------------------------------------------------------------------------
// Kernel A: out = hidden @ W^T + b, reshape to (B,S,9,128), split Q|K,
//           apply RoPE, scatter into qbuf/kbuf laid out [(b*9+h)][s][64].
// Grid: (8192/64, 1152/64) = (128, 18), block = 256 (8 waves).
// ---------------------------------------------------------------------------
__global__ __launch_bounds__(256) void gp_gemm_rope_kernel(
    const float* __restrict__ hidden,   // (B*S, H) row-major
    const float* __restrict__ W,        // (N_OUT, H) row-major
    const float* __restrict__ bias,     // (N_OUT,)
    float* __restrict__ qbuf,           // (B*NER, S, 64)
    float* __restrict__ kbuf)           // (B*NER, S, 64)
{
    __shared__ float lA[64 * LDS_STRIDE];
    __shared__ float lB[64 * LDS_STRIDE];

    const int m0   = blockIdx.x * 64;       // row tile in [0, B*S)
    const int n0   = blockIdx.y * 64;       // col tile in [0, N_OUT)
    const int t    = threadIdx.x;
    const int lane = t & 31;
    const int wave = t >> 5;
    const int ni   = wave & 3;              // N subtile (shared by both accs)
    const int mi0  = (wave >> 2) * 2;       // first M subtile
    const int lrow = lane & 15;
    const int koff = (lane >> 4) * 2;

    // Per-thread staging coordinates (4 float4 per thread per tile).
    v8f acc0 = {};
    v8f acc1 = {};

    for (int kb = 0; kb < H_DIM; kb += 64) {
        // Stage 64x64 tiles of hidden and W into LDS.
        #pragma unroll
        for (int q = 0; q < 4; ++q) {
            const int idx = t + 256 * q;
            const int r   = idx >> 4;
            const int c4  = (idx & 15) << 2;
            *(float4*)&lA[r * LDS_STRIDE + c4] =
                *(const float4*)&hidden[(size_t)(m0 + r) * H_DIM + kb + c4];
            *(float4*)&lB[r * LDS_STRIDE + c4] =
                *(const float4*)&W[(size_t)(n0 + r) * H_DIM + kb + c4];
        }
        __syncthreads();

        // Prefetch next K-block into cache while this block's WMMAs run.
        if (kb + 64 < H_DIM) {
            const int idx = t;
            const int r   = idx >> 4;
            const int c4  = (idx & 15) << 2;
            __builtin_prefetch(&hidden[(size_t)(m0 + r) * H_DIM + kb + 64 + c4], 0, 0);
            __builtin_prefetch(&W[(size_t)(n0 + r) * H_DIM + kb + 64 + c4], 0, 0);
        }

        const float* Ap0 = &lA[(mi0 * 16 + lrow)       * LDS_STRIDE + koff];
        const float* Ap1 = &lA[((mi0 + 1) * 16 + lrow) * LDS_STRIDE + koff];
        const float* Bp  = &lB[(ni * 16 + lrow)        * LDS_STRIDE + koff];

        #pragma unroll
        for (int kk = 0; kk < 64; kk += 4) {
            const v2f a0 = *(const v2f*)&Ap0[kk];
            const v2f a1 = *(const v2f*)&Ap1[kk];
            const v2f bm = *(const v2f*)&Bp[kk];
            acc0 = __builtin_amdgcn_wmma_f32_16x16x4_f32(
                false, a0, false, bm, (short)0, acc0, false, false);
            acc1 = __builtin_amdgcn_wmma_f32_16x16x4_f32(
                false, a1, false, bm, (short)0, acc1, false, false);
        }
        __syncthreads();
    }

    // ---- Epilogue: bias + RoPE + scatter to q/k buffers -------------------
    const int jcol   = n0 + ni * 16 + lrow;   // column in [0, 1152)
    const int head   = jcol >> 7;             // / 128
    const int within = jcol & 127;
    const bool is_q  = within < D_INNER;
    const int  dd    = is_q ? within : (within - D_INNER);
    const int  pairI = dd >> 1;
    const bool even  = (dd & 1) == 0;

    // inv_freq = 10000^(-2*pairI/64) = exp(pairI * (-2*ln(10000)/64))
    const float inv  = __expf((float)pairI * -0.28782313662425575f);
    const float bval = bias[jcol];
    float* __restrict__ dst = is_q ? qbuf : kbuf;

    const int rowAdd = (lane >> 4) * 8;

    #pragma unroll
    for (int sub = 0; sub < 2; ++sub) {
        const v8f acc = sub ? acc1 : acc0;
        const int mi  = mi0 + sub;
        #pragma unroll
        for (int r = 0; r < 8; ++r) {
            const float v = acc[r] + bval;
            const float p = __shfl_xor(v, 1, 32);   // RoPE partner column
            const int mg  = m0 + mi * 16 + r + rowAdd;
            const int s   = mg & (S_LEN - 1);
            const int b   = mg >> 9;                // / S_LEN
            const float ang = (float)s * inv;
            const float c   = __cosf(ang);
            const float sn  = __sinf(ang);
            const float out = even ? (v * c - p * sn) : (v * c + p * sn);
            dst[(((size_t)(b * NER + head)) * S_LEN + s) * D_INNER + dd] = out;
        }
    }
}

// ---------------------------------------------------------------------------
// Kernel B: logits[b,h,m,n] = (Q[b,h,m,:] . K[b,h,n,:]) masked & scaled.
// Grid: (B*NER, S/64, S/64) = (144, 8, 8), block = 256 (8 waves).
// qbuf/kbuf (38 MB total) live in the 192 MB L2, so the 8x tile re-reads are
// L2 hits. The 151 MB logits write is the bottleneck, so the masked tile is
// restaged through LDS and written back as contiguous float4 rows (256 B per
// 16-lane group) instead of the split 64 B segments of the raw D layout.
// ---------------------------------------------------------------------------
__global__ __launch_bounds__(256) void gp_logits_kernel(
    const float* __restrict__ qbuf,
    const float* __restrict__ kbuf,
    const float* __restrict__ attn_mask,  // (B, S)
    float* __restrict__ out)              // (B, NER, S, S)
{
    __shared__ float lQ[64 * LDS_STRIDE];
    __shared__ float lK[64 * LDS_STRIDE];

    const int bh    = blockIdx.x;           // b*NER + h
    const int b     = bh / NER;
    const int mtile = blockIdx.y * 64;
    const int ntile = blockIdx.z * 64;
    const int t     = threadIdx.x;
    const int lane  = t & 31;
    const int wave  = t >> 5;
    const int ni    = wave & 3;
    const int mi0   = (wave >> 2) * 2;
    const int lrow  = lane & 15;
    const int koff  = (lane >> 4) * 2;

    // Stage Q (rows mtile..+63) and K (rows ntile..+63), 64 dims each.
    #pragma unroll
    for (int q = 0; q < 4; ++q) {
        const int idx = t + 256 * q;
        const int r   = idx >> 4;
        const int c4  = (idx & 15) << 2;
        *(float4*)&lQ[r * LDS_STRIDE + c4] =
            *(const float4*)&qbuf[((size_t)bh * S_LEN + mtile + r) * D_INNER + c4];
        *(float4*)&lK[r * LDS_STRIDE + c4] =
            *(const float4*)&kbuf[((size_t)bh * S_LEN + ntile + r) * D_INNER + c4];
    }
    __syncthreads();

    v8f acc0 = {};
    v8f acc1 = {};
    const float* Ap0 = &lQ[(mi0 * 16 + lrow)       * LDS_STRIDE + koff];
    const float* Ap1 = &lQ[((mi0 + 1) * 16 + lrow) * LDS_STRIDE + koff];
    const float* Bp  = &lK[(ni * 16 + lrow)        * LDS_STRIDE + koff];

    #pragma unroll
    for (int kk = 0; kk < D_INNER; kk += 4) {
        const v2f a0 = *(const v2f*)&Ap0[kk];
        const v2f a1 = *(const v2f*)&Ap1[kk];
        const v2f bm = *(const v2f*)&Bp[kk];
        acc0 = __builtin_amdgcn_wmma_f32_16x16x4_f32(
            false, a0, false, bm, (short)0, acc0, false, false);
        acc1 = __builtin_amdgcn_wmma_f32_16x16x4_f32(
            false, a1, false, bm, (short)0, acc1, false, false);
    }

    // ---- Epilogue: pad mask, causal mask, 1/sqrt(d), LDS-restage, store ---
    const int   ng  = ntile + ni * 16 + lrow;           // key position
    const float pad = attn_mask[(size_t)b * S_LEN + ng];
    const float padNeg = (1.0f - pad) * NEG_BIG;
    const int rowAdd = (lane >> 4) * 8;
    const int ncol   = ni * 16 + lrow;                  // column within tile

    __syncthreads();   // Q tile fully consumed; reuse lQ as the output stage.

    #pragma unroll
    for (int sub = 0; sub < 2; ++sub) {
        const v8f acc = sub ? acc1 : acc0;
        const int mi  = mi0 + sub;
        #pragma unroll
        for (int r = 0; r < 8; ++r) {
            const int mrow = mi * 16 + r + rowAdd;        // row within tile
            const int mg   = mtile + mrow;                // query position
            float v = acc[r] * pad - padNeg;
            if (mg > ng) v -= NEG_BIG;                    // tril(-1) causal mask
            lQ[mrow * LDS_STRIDE + ncol] = v * 0.125f;    // / sqrt(64)
        }
    }
    __syncthreads();

    // Coalesced writeback: 16 consecutive lanes cover 256 B of one row.
    #pragma unroll
    for (int q = 0; q < 4; ++q) {
        const int idx = t + 256 * q;
        const int r   = idx >> 4;
        const int c4  = (idx & 15) << 2;
        const float4 vv = *(const float4*)&lQ[r * LDS_STRIDE + c4];
        *(float4*)&out[((size_t)bh * S_LEN + mtile + r) * S_LEN + ntile + c4] = vv;
    }
}

// ---------------------------------------------------------------------------
extern "C" void kernel_launch(void* const* d_in, const int* in_sizes, int n_in,
                              void* d_out, int out_size, void* d_ws, size_t ws_size,
                              hipStream_t stream) {
    const float* hidden = (const float*)d_in[0];   // (16,512,768)
    const float* mask   = (const float*)d_in[1];   // (16,512)
    const float* W      = (const float*)d_in[2];   // (1152,768)
    const float* bias   = (const float*)d_in[3];   // (1152,)
    float* out          = (float*)d_out;           // (16,9,512,512)

    const size_t qk_elems = (size_t)B_SZ * NER * S_LEN * D_INNER; // 4.7M floats
    float* qbuf = (float*)d_ws;
    float* kbuf = qbuf + qk_elems;

    dim3 gridA((B_SZ * S_LEN) / 64, N_OUT / 64);   // (128, 18)
    gp_gemm_rope_kernel<<<gridA, 256, 0, stream>>>(hidden, W, bias, qbuf, kbuf);

    dim3 gridB(B_SZ * NER, S_LEN / 64, S_LEN / 64); // (144, 8, 8)
    gp_logits_kernel<<<gridB, 256, 0, stream>>>(qbuf, kbuf, mask, out);
}